// MultiHeadGraphAttention_38628935860830
// MI455X (gfx1250) — compile-verified
//
#include <hip/hip_runtime.h>
#include <hip/hip_bf16.h>

// Problem constants (fixed by the reference)
#define NN 50000
#define EE 600000
#define DD 128
#define HH 8
#define DHD 16
#define EDD 64
// SCALE = DH^-0.5 = 0.25f

typedef __attribute__((ext_vector_type(2))) float    v2f;
typedef __attribute__((ext_vector_type(8))) float    v8f;
typedef __attribute__((ext_vector_type(4))) unsigned u32x4;
typedef __attribute__((ext_vector_type(4))) int      i32x4;
typedef __attribute__((ext_vector_type(8))) int      i32x8;

// ---------------------------------------------------------------------------
// TDM: one 2-D TENSOR_LOAD_TO_LDS of a rows x cols fp32 panel (row stride =
// cols) into LDS at ldsOff. Descriptor packing per CDNA5 ISA 8.3/8.4.
// All fields wave-uniform (SGPR descriptor groups). Groups 2/3 zero: 2-D
// tensor, tile_dim2..4 = 0 (unused).
// ---------------------------------------------------------------------------
__device__ __forceinline__ void tdm_load_panel_f32(const float* gsrc,
                                                   unsigned ldsOff,
                                                   int rows, int cols) {
    unsigned long long ga = (unsigned long long)(uintptr_t)gsrc;
    u32x4 g0;
    g0[0] = 1u;                                      // count=1, user mode
    g0[1] = ldsOff;                                  // lds_addr (bytes)
    g0[2] = (unsigned)(ga & 0xffffffffu);            // global_addr[31:0]
    g0[3] = (unsigned)((ga >> 32) & 0x01ffffffu)     // global_addr[56:32]
          | (2u << 30);                              // type = 2 ("image")
    i32x8 g1;
    g1[0] = (int)(2u << 16);                         // data_size=2 (4B), mask=0
    g1[1] = (int)(((unsigned)cols & 0xffffu) << 16); // tensor_dim0[15:0]
    g1[2] = (int)((((unsigned)cols >> 16) & 0xffffu)
          | (((unsigned)rows & 0xffffu) << 16));     // tdim0 hi | tensor_dim1 lo
    g1[3] = (int)((((unsigned)rows >> 16) & 0xffffu)
          | (((unsigned)cols & 0xffffu) << 16));     // tdim1 hi | tile_dim0=cols
    g1[4] = (int)((unsigned)rows & 0xffffu);         // tile_dim1=rows, tile_dim2=0
    g1[5] = (int)(unsigned)cols;                     // tensor_dim0_stride lo
    g1[6] = 0;                                       // stride hi | dim1_stride lo
    g1[7] = 0;
    i32x4 z4 = {0, 0, 0, 0};
#if __clang_major__ >= 23
    i32x8 z8 = {0, 0, 0, 0, 0, 0, 0, 0};
    __builtin_amdgcn_tensor_load_to_lds(g0, g1, z4, z4, z8, 0);
#else
    __builtin_amdgcn_tensor_load_to_lds(g0, g1, z4, z4, 0);
#endif
}

// ---------------------------------------------------------------------------
// fp32 WMMA GEMM, 2x2 register-blocked: each wave32 computes a 32x32 output
// tile of C[M x Nc] = A[M x K] @ B[K x Nc] + bias[Nc], K stepped by 4 with
// V_WMMA_F32_16X16X4_F32. The full K x Nc B panel (<= 64KB) is staged into
// LDS once per workgroup by the Tensor Data Mover; fragments then come from
// ds_load. Fragment layouts per CDNA5 ISA 7.12.2:
//   A 16x4 : lane l, vgpr v -> (M = l%16, K = 2*(l/16)+v)   (contiguous f2/lane)
//   B 4x16 : lane l, vgpr v -> (K = v + 2*(l/16), N = l%16)
//   C 16x16: lane l, vgpr j -> (M = j + 8*(l/16), N = l%16)
// ---------------------------------------------------------------------------
__global__ __launch_bounds__(256)
void mhga_gemm_wmma_f32(const float* __restrict__ A, const float* __restrict__ B,
                        const float* __restrict__ bias, float* __restrict__ C,
                        int M, int K, int Nc) {
    __shared__ float shB[128 * 128];   // 64 KB: K x Nc panel (K <= 128, Nc = 128)

    const int lane = threadIdx.x & 31;
    const int wave = threadIdx.x >> 5;
    const int half = lane >> 4;    // 0 or 1
    const int l16  = lane & 15;

    // Stage B panel via TDM (one op per workgroup), then barrier.
    if (wave == 0) {
        tdm_load_panel_f32(B, (unsigned)(uintptr_t)(void*)shB, K, Nc);
        __builtin_amdgcn_s_wait_tensorcnt(0);
    }
    __syncthreads();

    const int nt2 = Nc >> 5;               // 32-wide col tiles
    const int mt2 = (M + 31) >> 5;         // 32-tall row tiles
    const int tile = blockIdx.x * 8 + wave;  // uniform per wave
    if (tile >= mt2 * nt2) return;           // whole wave exits (after barrier)
    const int mt = tile / nt2;
    const int nt = tile - mt * nt2;
    const int col0 = nt * 32 + l16;
    const int col1 = col0 + 16;

    int r0 = mt * 32 + l16;       if (r0 >= M) r0 = M - 1;   // clamp partial tile
    int r1 = mt * 32 + 16 + l16;  if (r1 >= M) r1 = M - 1;

    const float bv0 = bias[col0];
    const float bv1 = bias[col1];
    v8f acc00, acc01, acc10, acc11;
#pragma unroll
    for (int j = 0; j < 8; ++j) {
        acc00[j] = bv0; acc01[j] = bv1;
        acc10[j] = bv0; acc11[j] = bv1;
    }

    const float* Ap0 = A + (size_t)r0 * K + 2 * half;  // k = k0 + 2*half + {0,1}
    const float* Ap1 = A + (size_t)r1 * K + 2 * half;
    const float* Bs0 = shB + (size_t)(2 * half) * Nc + col0;
    const float* Bs1 = shB + (size_t)(2 * half) * Nc + col1;

    for (int k0 = 0; k0 < K; k0 += 4) {
        if ((k0 & 31) == 0 && (k0 + 32) < K) {
            // gfx1250 global_prefetch_b8: pull next A cachelines toward L0/L2
            __builtin_prefetch(Ap0 + k0 + 32, 0, 1);
            __builtin_prefetch(Ap1 + k0 + 32, 0, 1);
        }
        v2f a0 = *(const v2f*)(Ap0 + k0);              // 8B aligned (even offset)
        v2f a1 = *(const v2f*)(Ap1 + k0);
        v2f b0, b1;                                    // ds_load_b32 x4
        b0.x = Bs0[(size_t)k0 * Nc];
        b0.y = Bs0[(size_t)(k0 + 1) * Nc];
        b1.x = Bs1[(size_t)k0 * Nc];
        b1.y = Bs1[(size_t)(k0 + 1) * Nc];
        acc00 = __builtin_amdgcn_wmma_f32_16x16x4_f32(false, a0, false, b0,
                                                      (short)0, acc00, false, false);
        acc01 = __builtin_amdgcn_wmma_f32_16x16x4_f32(false, a0, false, b1,
                                                      (short)0, acc01, false, false);
        acc10 = __builtin_amdgcn_wmma_f32_16x16x4_f32(false, a1, false, b0,
                                                      (short)0, acc10, false, false);
        acc11 = __builtin_amdgcn_wmma_f32_16x16x4_f32(false, a1, false, b1,
                                                      (short)0, acc11, false, false);
    }

#pragma unroll
    for (int j = 0; j < 8; ++j) {
        const int ra = mt * 32 + half * 8 + j;       // rows of acc00/acc01
        const int rb = ra + 16;                      // rows of acc10/acc11
        if (ra < M) {
            C[(size_t)ra * Nc + col0] = acc00[j];
            C[(size_t)ra * Nc + col1] = acc01[j];
        }
        if (rb < M) {
            C[(size_t)rb * Nc + col0] = acc10[j];
            C[(size_t)rb * Nc + col1] = acc11[j];
        }
    }
}

// ---------------------------------------------------------------------------
// Sortable-uint encoding for float atomicMax
// ---------------------------------------------------------------------------
__device__ __forceinline__ unsigned f32_sortable(float f) {
    unsigned u = __float_as_uint(f);
    return (u & 0x80000000u) ? ~u : (u | 0x80000000u);
}
__device__ __forceinline__ float sortable_f32(unsigned u) {
    return (u & 0x80000000u) ? __uint_as_float(u & 0x7fffffffu)
                             : __uint_as_float(~u);
}

// ---------------------------------------------------------------------------
// Per-(edge, head) logits: q[dst] . (k[src] + e_proj[edge]) * 0.25
// Segment max via sortable-uint atomicMax.
// ---------------------------------------------------------------------------
__global__ __launch_bounds__(256)
void mhga_edge_logits(const float* __restrict__ Q, const float* __restrict__ Kf,
                      const float* __restrict__ Ep,
                      const int* __restrict__ src, const int* __restrict__ dst,
                      float* __restrict__ logits, unsigned* __restrict__ mmax) {
    int idx = blockIdx.x * 256 + threadIdx.x;
    if (idx >= EE * HH) return;
    int ei = idx >> 3;       // / H
    int h  = idx & 7;        // % H
    int s = src[ei], d = dst[ei];
    const float4* qp = (const float4*)(Q  + (size_t)d  * DD + h * DHD);
    const float4* kp = (const float4*)(Kf + (size_t)s  * DD + h * DHD);
    const float4* ep = (const float4*)(Ep + (size_t)ei * DD + h * DHD);
    float acc = 0.f;
#pragma unroll
    for (int i = 0; i < 4; ++i) {
        float4 q = qp[i], k = kp[i], e = ep[i];
        acc += q.x * (k.x + e.x) + q.y * (k.y + e.y)
             + q.z * (k.z + e.z) + q.w * (k.w + e.w);
    }
    acc *= 0.25f;  // DH^-0.5
    logits[idx] = acc;
    atomicMax(&mmax[(size_t)d * HH + h], f32_sortable(acc));
}

// ---------------------------------------------------------------------------
// ex = exp(logit - max); segment-sum denominator
// ---------------------------------------------------------------------------
__global__ __launch_bounds__(256)
void mhga_edge_exp(const int* __restrict__ dst, const unsigned* __restrict__ mmax,
                   float* __restrict__ logits, float* __restrict__ denom) {
    int idx = blockIdx.x * 256 + threadIdx.x;
    if (idx >= EE * HH) return;
    int ei = idx >> 3;
    int h  = idx & 7;
    int d = dst[ei];
    float m = sortable_f32(mmax[(size_t)d * HH + h]);
    float ex = __expf(logits[idx] - m);
    logits[idx] = ex;                 // reuse buffer as ex
    atomicAdd(&denom[(size_t)d * HH + h], ex);
}

// ---------------------------------------------------------------------------
// Per-(edge, 4 channels): accum[dst, c..c+3] += (ex/denom) * V[src, c..c+3]
// One float4 V load + one index fetch per 4 atomics.
// ---------------------------------------------------------------------------
__global__ __launch_bounds__(256)
void mhga_edge_scatter(const float* __restrict__ V, const float* __restrict__ ex,
                       const float* __restrict__ denom,
                       const int* __restrict__ src, const int* __restrict__ dst,
                       float* __restrict__ accum) {
    long long idx = (long long)blockIdx.x * 256 + threadIdx.x;
    if (idx >= (long long)EE * (DD / 4)) return;
    int ei = (int)(idx >> 5);          // / 32 quads per edge
    int c  = ((int)idx & 31) * 4;      // channel base
    int h  = c >> 4;                   // head
    int s = src[ei], d = dst[ei];
    float a = ex[(size_t)ei * HH + h] / denom[(size_t)d * HH + h];
    float4 v = *(const float4*)(V + (size_t)s * DD + c);
    float* ap = accum + (size_t)d * DD + c;
    atomicAdd(ap + 0, a * v.x);
    atomicAdd(ap + 1, a * v.y);
    atomicAdd(ap + 2, a * v.z);
    atomicAdd(ap + 3, a * v.w);
}

// ---------------------------------------------------------------------------
extern "C" void kernel_launch(void* const* d_in, const int* in_sizes, int n_in,
                              void* d_out, int out_size, void* d_ws, size_t ws_size,
                              hipStream_t stream) {
    const float* x         = (const float*)d_in[0];
    const float* edge_attr = (const float*)d_in[1];
    const float* Wq = (const float*)d_in[2];
    const float* bq = (const float*)d_in[3];
    const float* Wk = (const float*)d_in[4];
    const float* bk = (const float*)d_in[5];
    const float* Wv = (const float*)d_in[6];
    const float* bv = (const float*)d_in[7];
    const float* We = (const float*)d_in[8];
    const float* be = (const float*)d_in[9];
    const float* Wo = (const float*)d_in[10];
    const float* bo = (const float*)d_in[11];
    const int* edge_index = (const int*)d_in[12];
    const int* src = edge_index;        // edge_index[0, :]
    const int* dst = edge_index + EE;   // edge_index[1, :]
    float* out = (float*)d_out;

    // Workspace layout (fp32 words)
    float* ws = (float*)d_ws;
    float*    Q      = ws;                                   // N*D
    float*    Kp     = Q  + (size_t)NN * DD;                 // N*D
    float*    Vp     = Kp + (size_t)NN * DD;                 // N*D
    float*    Ep     = Vp + (size_t)NN * DD;                 // E*D
    float*    logits = Ep + (size_t)EE * DD;                 // E*H (reused as ex)
    unsigned* mmax   = (unsigned*)(logits + (size_t)EE * HH);// N*H
    float*    denom  = (float*)(mmax + (size_t)NN * HH);     // N*H
    float*    accum  = denom + (size_t)NN * HH;              // N*D

    // Zero {mmax, denom, accum} (contiguous) every call: sortable-uint 0 == -inf
    hipMemsetAsync(mmax, 0,
                   ((size_t)NN * HH * 2 + (size_t)NN * DD) * sizeof(float),
                   stream);

    const dim3 blk(256);
    // 32x32 tiles, 8 waves (tiles) per block
    const int tiles_n = ((NN + 31) / 32) * (DD / 32);
    const int grid_n  = (tiles_n + 7) / 8;
    const int tiles_e = ((EE + 31) / 32) * (DD / 32);
    const int grid_e  = (tiles_e + 7) / 8;

    // 1) Q, K, V projections
    mhga_gemm_wmma_f32<<<grid_n, blk, 0, stream>>>(x, Wq, bq, Q,  NN, DD, DD);
    mhga_gemm_wmma_f32<<<grid_n, blk, 0, stream>>>(x, Wk, bk, Kp, NN, DD, DD);
    mhga_gemm_wmma_f32<<<grid_n, blk, 0, stream>>>(x, Wv, bv, Vp, NN, DD, DD);
    // 2) Edge projection
    mhga_gemm_wmma_f32<<<grid_e, blk, 0, stream>>>(edge_attr, We, be, Ep, EE, EDD, DD);
    // 3) Logits + segment max
    const int grid_eh = (EE * HH + 255) / 256;
    mhga_edge_logits<<<grid_eh, blk, 0, stream>>>(Q, Kp, Ep, src, dst, logits, mmax);
    // 4) exp + segment sum
    mhga_edge_exp<<<grid_eh, blk, 0, stream>>>(dst, mmax, logits, denom);
    // 5) Weighted scatter of V (4 channels per thread)
    const long long eq = (long long)EE * (DD / 4);
    const int grid_ec = (int)((eq + 255) / 256);
    mhga_edge_scatter<<<grid_ec, blk, 0, stream>>>(Vp, logits, denom, src, dst, accum);
    // 6) Output projection -> d_out
    mhga_gemm_wmma_f32<<<grid_n, blk, 0, stream>>>(accum, Wo, bo, out, NN, DD, DD);
}